// GATLayer_22041772163468
// MI455X (gfx1250) — compile-verified
//
#include <hip/hip_runtime.h>
#include <stdint.h>

// ---------------------------------------------------------------------------
// GATConv forward for MI455X (gfx1250, wave32).
//   h = x @ W            -> fp32 WMMA 16x16x4, LDS-staged A (async copy),
//                           64-row M-blocks for 4x B-fragment reuse
//   a_src/a_dst          -> wave32 shuffle-reduced dot products
//   edge softmax+scatter -> 3 edge passes with L2-resident float atomics
// Sizes: N=50000, E=800000 (+N self loops), IN=256, H=4, C=32 (HC=128).
// ---------------------------------------------------------------------------

#define IN_CH   256
#define HC      128   // HEADS * OUT_CH
#define HEADS   4
#define OUT_C   32
#define BM      64                 // rows of x per block (4 WMMA m-tiles)
#define LDS_STR (IN_CH + 4)        // row stride: 16B-aligned rows, conflict-free

typedef __attribute__((ext_vector_type(2))) float v2f;
typedef __attribute__((ext_vector_type(8))) float v8f;
typedef int v4i __attribute__((vector_size(16)));   // async-copy operand type

#if defined(__has_builtin)
#if __has_builtin(__builtin_amdgcn_global_load_async_to_lds_b128) && \
    __has_builtin(__builtin_amdgcn_s_wait_asynccnt)
#define USE_ASYNC_LDS 1
#endif
#endif

// ---------------- GEMM: h[N,128] = x[N,256] @ W[256,128] -------------------
// Block: 256 threads (8 waves) -> 64 rows x 128 cols. Wave w owns n-tile w;
// loops over 4 m-subtiles sharing one B fragment per k-step.
// fp32 A (16x4): lane L holds row L%16; VGPR0/1 = K 2*(L>=16)+{0,1}.
// fp32 B (4x16): lane L holds col L%16; VGPR0/1 = K 2*(L>=16)+{0,1}.
// fp32 C/D:      lane L, reg v -> row v + 8*(L>=16), col L%16.
__global__ __launch_bounds__(256) void gat_gemm_wmma(
    const float* __restrict__ X, const float* __restrict__ W,
    float* __restrict__ H, int n_nodes) {
  __shared__ float xs[BM * LDS_STR];   // ~66.6 KB of the 320 KB WGP LDS

  const int tid = threadIdx.x;
  const int m0  = blockIdx.x * BM;

  // ---- stage x tile [BM x 256] into LDS (rows clamped for the tail block)
#pragma unroll
  for (int i = 0; i < (BM * (IN_CH / 4)) / 256; ++i) {   // 16 x float4/thread
    const int idx  = tid + i * 256;
    const int row  = idx >> 6;            // /64 float4 per row
    const int c4   = idx & 63;
    const int grow = min(m0 + row, n_nodes - 1);
    const float* gp = X + (size_t)grow * IN_CH + c4 * 4;
    float*       lp = &xs[row * LDS_STR + c4 * 4];
#if USE_ASYNC_LDS
    __builtin_amdgcn_global_load_async_to_lds_b128((v4i*)gp, (v4i*)lp, 0, 0);
#else
    *(float4*)lp = *(const float4*)gp;
#endif
  }
#if USE_ASYNC_LDS
  __builtin_amdgcn_s_wait_asynccnt(0);
#endif
  __syncthreads();

  const int lane = tid & 31;
  const int wave = tid >> 5;
  const int hi   = lane >> 4;
  const int l15  = lane & 15;
  const int n    = wave * 16 + l15;      // global output column

  v8f acc[4] = {};
  for (int k = 0; k < IN_CH; k += 4) {
    const int kk = k + 2 * hi;
    v2f b;
    b.x = W[(size_t)kk * HC + n];
    b.y = W[(size_t)(kk + 1) * HC + n];
#pragma unroll
    for (int mt = 0; mt < 4; ++mt) {
      const v2f a = *(const v2f*)&xs[(mt * 16 + l15) * LDS_STR + kk];
      acc[mt] = __builtin_amdgcn_wmma_f32_16x16x4_f32(
          /*neg_a=*/false, a, /*neg_b=*/false, b,
          /*c_mod=*/(short)0, acc[mt], /*reuse_a=*/false, /*reuse_b=*/false);
    }
  }
#pragma unroll
  for (int mt = 0; mt < 4; ++mt)
#pragma unroll
    for (int v = 0; v < 8; ++v) {
      const int row = m0 + mt * 16 + v + 8 * hi;
      if (row < n_nodes) H[(size_t)row * HC + n] = acc[mt][v];
    }
}

// ------------- a_src/a_dst: wave per (node, head), 32-lane dot -------------
__global__ __launch_bounds__(256) void gat_att(
    const float* __restrict__ H, const float* __restrict__ att_src,
    const float* __restrict__ att_dst, float* __restrict__ a_src,
    float* __restrict__ a_dst, int n_nodes) {
  const int wid  = (int)((blockIdx.x * blockDim.x + threadIdx.x) >> 5);
  const int lane = threadIdx.x & 31;
  if (wid >= n_nodes * HEADS) return;
  const int node = wid >> 2;
  const int head = wid & 3;
  const float hv = H[(size_t)node * HC + head * OUT_C + lane];
  float s = hv * att_src[head * OUT_C + lane];
  float d = hv * att_dst[head * OUT_C + lane];
#pragma unroll
  for (int off = 16; off > 0; off >>= 1) {
    s += __shfl_down(s, off, 32);
    d += __shfl_down(d, off, 32);
  }
  if (lane == 0) { a_src[wid] = s; a_dst[wid] = d; }
}

// -------------------------- init out/emax/denom ----------------------------
__global__ void gat_init(float* __restrict__ out, const float* __restrict__ bias,
                         float* __restrict__ emax, float* __restrict__ denom,
                         int n_nodes) {
  const int i = blockIdx.x * blockDim.x + threadIdx.x;
  if (i < n_nodes * HC) out[i] = bias[i & (HC - 1)];
  if (i < n_nodes * HEADS) { emax[i] = -__builtin_huge_valf(); denom[i] = 0.f; }
}

// sign-aware float atomic max (lowers to global_atomic_max_i32/min_u32 in L2)
__device__ inline void atomicMaxFloat(float* addr, float value) {
  if (value >= 0.f)
    atomicMax((int*)addr, __float_as_int(value));
  else
    atomicMin((unsigned int*)addr, __float_as_uint(value));
}

__device__ inline void edge_endpoints(const long long* __restrict__ ei, int e,
                                      int E, int* s, int* d) {
  if (e < E) { *s = (int)ei[e]; *d = (int)ei[E + e]; }
  else       { *s = *d = e - E; }   // self loop
}

// ---------------- pass 1: logits + per-dst running max ---------------------
__global__ void gat_logits(const long long* __restrict__ ei,
                           const float* __restrict__ a_src,
                           const float* __restrict__ a_dst,
                           float* __restrict__ emax, float* __restrict__ ebuf,
                           int E, int n_nodes) {
  const int i = blockIdx.x * blockDim.x + threadIdx.x;
  if (i >= (E + n_nodes) * HEADS) return;
  const int e = i >> 2, h = i & 3;
  int s, d; edge_endpoints(ei, e, E, &s, &d);
  float v = a_src[s * HEADS + h] + a_dst[d * HEADS + h];
  v = (v > 0.f) ? v : 0.2f * v;            // leaky_relu(0.2)
  ebuf[i] = v;
  atomicMaxFloat(&emax[d * HEADS + h], v);
}

// ---------------- pass 2: exp(e - max) + denom sum -------------------------
__global__ void gat_expsum(const long long* __restrict__ ei,
                           const float* __restrict__ emax,
                           float* __restrict__ denom, float* __restrict__ ebuf,
                           int E, int n_nodes) {
  const int i = blockIdx.x * blockDim.x + threadIdx.x;
  if (i >= (E + n_nodes) * HEADS) return;
  const int e = i >> 2, h = i & 3;
  int s, d; edge_endpoints(ei, e, E, &s, &d);
  const float v = __expf(ebuf[i] - emax[d * HEADS + h]);
  ebuf[i] = v;
  atomicAdd(&denom[d * HEADS + h], v);
}

// ---------------- pass 3: alpha-weighted scatter of h[src] -----------------
__global__ void gat_scatter(const long long* __restrict__ ei,
                            const float* __restrict__ H,
                            const float* __restrict__ ebuf,
                            const float* __restrict__ denom,
                            float* __restrict__ out, int E, int n_nodes) {
  const long long gid = (long long)blockIdx.x * blockDim.x + threadIdx.x;
  if (gid >= (long long)(E + n_nodes) * HC) return;
  const int e = (int)(gid >> 7);
  const int c = (int)(gid & (HC - 1));
  const int h = c >> 5;
  int s, d; edge_endpoints(ei, e, E, &s, &d);
  const float alpha = ebuf[e * HEADS + h] / (denom[d * HEADS + h] + 1e-16f);
  atomicAdd(&out[(size_t)d * HC + c], alpha * H[(size_t)s * HC + c]);
}

// ---------------------------------------------------------------------------
extern "C" void kernel_launch(void* const* d_in, const int* in_sizes, int n_in,
                              void* d_out, int out_size, void* d_ws, size_t ws_size,
                              hipStream_t stream) {
  const float*     x       = (const float*)d_in[0];
  const long long* ei      = (const long long*)d_in[1];   // int64 per reference
  const float*     W       = (const float*)d_in[2];
  const float*     att_src = (const float*)d_in[3];
  const float*     att_dst = (const float*)d_in[4];
  const float*     bias    = (const float*)d_in[5];
  float*           out     = (float*)d_out;

  const int N = in_sizes[0] / IN_CH;   // 50000
  const int E = in_sizes[1] / 2;       // 800000
  const int Etot = E + N;

  // workspace carve-up (floats): h | a_src | a_dst | emax | denom | ebuf
  float* ws    = (float*)d_ws;
  float* Hbuf  = ws;                              // N*128
  float* a_src = Hbuf  + (size_t)N * HC;          // N*4
  float* a_dst = a_src + (size_t)N * HEADS;       // N*4
  float* emax  = a_dst + (size_t)N * HEADS;       // N*4
  float* denom = emax  + (size_t)N * HEADS;       // N*4
  float* ebuf  = denom + (size_t)N * HEADS;       // Etot*4

  const int B = 256;
  auto cdiv = [](long long a, long long b) { return (int)((a + b - 1) / b); };

  gat_init<<<cdiv((long long)N * HC, B), B, 0, stream>>>(out, bias, emax, denom, N);

  gat_gemm_wmma<<<cdiv(N, BM), B, 0, stream>>>(x, W, Hbuf, N);

  gat_att<<<cdiv((long long)N * HEADS * 32, B), B, 0, stream>>>(
      Hbuf, att_src, att_dst, a_src, a_dst, N);

  gat_logits<<<cdiv((long long)Etot * HEADS, B), B, 0, stream>>>(
      ei, a_src, a_dst, emax, ebuf, E, N);

  gat_expsum<<<cdiv((long long)Etot * HEADS, B), B, 0, stream>>>(
      ei, emax, denom, ebuf, E, N);

  gat_scatter<<<cdiv((long long)Etot * HC, B), B, 0, stream>>>(
      ei, Hbuf, ebuf, denom, out, E, N);
}